// RandomContrast_71734543777925
// MI455X (gfx1250) — compile-verified
//
#include <hip/hip_runtime.h>

// RandomContrast: out = (x - mean_{H,W}(x)) * c + mean, fp32
// B,C,H,W = 64,3,512,512  -> 192 planes of 1 MB each.
// Strategy: one 1024-thread block per (b,c) plane; fused reduce+apply so the
// second read of the plane hits L2 (reuse distance 1 MB << 192 MB L2).
// HBM traffic = 402 MB -> ~17 us roofline @ 23.3 TB/s.

#define CC 3
constexpr int kPlaneElems = 512 * 512;       // 262144
constexpr int kPlaneVec4  = kPlaneElems / 4; // 65536
constexpr int kThreads    = 1024;            // 32 waves (wave32)
constexpr int kIters      = kPlaneVec4 / kThreads; // 64

typedef float v2f __attribute__((ext_vector_type(2)));
typedef float v4f __attribute__((ext_vector_type(4)));
typedef float v8f __attribute__((ext_vector_type(8)));

// Full-wave (32-lane) fp32 sum using V_WMMA_F32_16X16X4_F32.
// A(16x4): lane L<16 holds {K0=s,K1=0} of row L; lane L>=16 holds {K2=s,K3=0}
// of row L-16. With B = ones(4x16): D[m,n] = s_m + s_{m+16} for every n.
// Per-lane vertical sum of the 8 C/D VGPRs gives half-sums; one shfl_xor(16)
// completes the reduction. EXEC must be all-1s (guaranteed at call sites).
__device__ __forceinline__ float wave_reduce_sum(float s) {
#if __has_builtin(__builtin_amdgcn_wmma_f32_16x16x4_f32)
    v2f a; a.x = s;    a.y = 0.0f;
    v2f b; b.x = 1.0f; b.y = 1.0f;
    v8f c = {};
    // 8 args: (neg_a, A, neg_b, B, c_mod, C, reuse_a, reuse_b)
    c = __builtin_amdgcn_wmma_f32_16x16x4_f32(
        /*neg_a=*/false, a, /*neg_b=*/false, b,
        /*c_mod=*/(short)0, c, /*reuse_a=*/false, /*reuse_b=*/false);
    float p = ((c[0] + c[1]) + (c[2] + c[3])) + ((c[4] + c[5]) + (c[6] + c[7]));
    p += __shfl_xor(p, 16, 32);
    return p;
#else
    #pragma unroll
    for (int off = 16; off > 0; off >>= 1) s += __shfl_xor(s, off, 32);
    return s;
#endif
}

__global__ __launch_bounds__(kThreads, 1)
void RandomContrast_71734543777925_kernel(const float* __restrict__ x,
                                          const float* __restrict__ contrast,
                                          float* __restrict__ out) {
    __shared__ float sdata[33];

    const int plane = blockIdx.x;             // 0..191
    const int batch = plane / CC;
    const size_t base4 = (size_t)plane * kPlaneVec4;
    const v4f* __restrict__ x4 = (const v4f*)x + base4;
    v4f* __restrict__ o4 = (v4f*)out + base4;
    const int tid = threadIdx.x;

    // ---- Pass 1: plane sum (regular loads: keep the plane cache-resident) ----
    v4f acc = {0.0f, 0.0f, 0.0f, 0.0f};
    #pragma unroll 4
    for (int i = 0; i < kIters; ++i) {
        v4f v = x4[tid + i * kThreads];
        acc += v;
    }
    float s = (acc.x + acc.y) + (acc.z + acc.w);
    s = wave_reduce_sum(s);

    const int wave = tid >> 5;
    const int lane = tid & 31;
    if (lane == 0) sdata[wave] = s;
    __syncthreads();

    if (tid < 32) {                    // exactly wave 0: EXEC all-1s inside
        float t = sdata[tid];          // 32 wave partials
        t = wave_reduce_sum(t);
        if (tid == 0) sdata[32] = t * (1.0f / (float)kPlaneElems);
    }
    __syncthreads();

    const float m = sdata[32];
    const float cst = contrast[batch];          // uniform -> scalar load
    const float addend = m * (1.0f - cst);      // out = x*c + m*(1-c)

    // ---- Pass 2: apply (last-use loads + streaming stores: TH=NT) ----
    #pragma unroll 4
    for (int i = 0; i < kIters; ++i) {
        const int idx = tid + i * kThreads;
        v4f v = __builtin_nontemporal_load(&x4[idx]);   // hits L2 (1 MB reuse)
        v4f r = v * cst + addend;
        __builtin_nontemporal_store(r, &o4[idx]);
    }
}

extern "C" void kernel_launch(void* const* d_in, const int* in_sizes, int n_in,
                              void* d_out, int out_size, void* d_ws, size_t ws_size,
                              hipStream_t stream) {
    (void)in_sizes; (void)n_in; (void)d_ws; (void)ws_size; (void)out_size;
    const float* x        = (const float*)d_in[0];
    const float* contrast = (const float*)d_in[1];
    float* out            = (float*)d_out;

    const int nPlanes = 64 * CC; // B*C = 192
    RandomContrast_71734543777925_kernel<<<nPlanes, kThreads, 0, stream>>>(x, contrast, out);
}